// Simulator_25967372272162
// MI455X (gfx1250) — compile-verified
//
#include <hip/hip_runtime.h>
#include <hip/hip_bf16.h>

// CDNA5 / gfx1250: wave32, WMMA 16x16x4 f32 for the scan-combine matrix work.
typedef __attribute__((ext_vector_type(2))) float v2f;
typedef __attribute__((ext_vector_type(8))) float v8f;

#define DT_F 3600.0f
#define CHUNK_LEN 128          // steps per chunk (= 2^7)
#define NCHUNKS 16384          // 2^21 / 128 chunks
#define NPOW 21                // M^(2^s), s = 0..20

// ---------------------------------------------------------------------------
// Kernel 0: build M = I + dt*A and compute M^(2^s) by WMMA squaring (1 wave).
// A(16x4) layout: lane<16 holds {row l: K0,K1}; lane>=16 holds {row l: K2,K3}.
// B(4x16) layout: lane<16 holds {col l: K0,K1}; lane>=16 holds {col l: K2,K3}.
// D(16x16): vgpr r = row r, lanes 0..15 = cols 0..15.
// ---------------------------------------------------------------------------
__global__ void __launch_bounds__(32) k_init(const float* __restrict__ params,
                                             float* __restrict__ pw) {
  __shared__ float P[9];
  const int lane = threadIdx.x;
  if (lane == 0) {
    float Cai = params[0], Cwe = params[1], Cwi = params[2];
    float Re = params[3], Ri = params[4], Rw = params[5], Rg = params[6];
    float A00 = -1.f / Cai * (1.f / Rg + 1.f / Ri);
    float A02 = 1.f / (Cai * Ri);
    float A11 = -1.f / Cwe * (1.f / Re + 1.f / Rw);
    float A12 = 1.f / (Cwe * Rw);
    float A20 = 1.f / (Cwi * Ri);
    float A21 = 1.f / (Cwi * Rw);
    float A22 = -1.f / Cwi * (1.f / Rw + 1.f / Ri);
    P[0] = 1.f + DT_F * A00; P[1] = 0.f;             P[2] = DT_F * A02;
    P[3] = 0.f;              P[4] = 1.f + DT_F * A11; P[5] = DT_F * A12;
    P[6] = DT_F * A20;       P[7] = DT_F * A21;       P[8] = 1.f + DT_F * A22;
  }
  __syncthreads();
  if (lane < 9) pw[lane] = P[lane];
  __syncthreads();

  const int half = lane >> 4;
  const int l = lane & 15;
  for (int s = 1; s < NPOW; ++s) {
    v2f a; a[0] = 0.f; a[1] = 0.f;
    v2f b; b[0] = 0.f; b[1] = 0.f;
    if (l < 3) {
      if (half == 0) {
        a[0] = P[l * 3 + 0]; a[1] = P[l * 3 + 1];   // A rows, K=0,1
        b[0] = P[0 * 3 + l]; b[1] = P[1 * 3 + l];   // B cols, K=0,1
      } else {
        a[0] = P[l * 3 + 2];                        // A rows, K=2 (K=3 pad 0)
        b[0] = P[2 * 3 + l];                        // B cols, K=2 (K=3 pad 0)
      }
    }
    v8f c = {0.f, 0.f, 0.f, 0.f, 0.f, 0.f, 0.f, 0.f};
    v8f d = __builtin_amdgcn_wmma_f32_16x16x4_f32(false, a, false, b,
                                                  (short)0, c, false, false);
    __syncthreads();
    if (lane < 3) {  // lane = column index; d[r] = row r of that column
      P[0 * 3 + lane] = d[0];
      P[1 * 3 + lane] = d[1];
      P[2 * 3 + lane] = d[2];
    }
    __syncthreads();
    if (lane < 9) pw[s * 9 + lane] = P[lane];
    __syncthreads();
  }
}

// ---------------------------------------------------------------------------
// Kernel 1: per-chunk reduction. One chunk of 128 steps per lane:
//   v <- M v + c_j,  c_j = dt * B d_j.  Streams d (40 MB) once.
// ---------------------------------------------------------------------------
__global__ void __launch_bounds__(256) k_reduce(const float* __restrict__ d,
                                                const float* __restrict__ pw,
                                                const float* __restrict__ params,
                                                float* __restrict__ vout,
                                                int nsteps) {
  const int k = blockIdx.x * blockDim.x + threadIdx.x;
  if (k >= NCHUNKS) return;
  float M[9];
#pragma unroll
  for (int i = 0; i < 9; ++i) M[i] = pw[i];
  const float Cai = params[0], Cwe = params[1], Cwi = params[2];
  const float Re = params[3], Rw = params[5], Rg = params[6];
  const float b00 = DT_F / (Cai * Rg), b01 = DT_F / Cai;
  const float b10 = DT_F / (Cwe * Re), b13 = DT_F / Cwe;
  const float b24 = DT_F / Cwi;

  float v0 = 0.f, v1 = 0.f, v2 = 0.f;
  const long long base = (long long)k * CHUNK_LEN;
  for (int s = 0; s < CHUNK_LEN; ++s) {
    const long long j = base + s;
    if (j >= nsteps) break;
    const float* dr = d + (size_t)j * 5;
    __builtin_prefetch(dr + 40, 0, 0);  // ~8 rows ahead -> global_prefetch
    const float d0 = dr[0], d1 = dr[1], d2 = dr[2], d3 = dr[3], d4 = dr[4];
    const float c0 = fmaf(b00, d0, b01 * (d1 + d2));
    const float c1 = fmaf(b10, d0, b13 * d3);
    const float c2 = b24 * d4;
    const float n0 = fmaf(M[0], v0, fmaf(M[1], v1, fmaf(M[2], v2, c0)));
    const float n1 = fmaf(M[3], v0, fmaf(M[4], v1, fmaf(M[5], v2, c1)));
    const float n2 = fmaf(M[6], v0, fmaf(M[7], v1, fmaf(M[8], v2, c2)));
    v0 = n0; v1 = n1; v2 = n2;
  }
  vout[k * 3 + 0] = v0;
  vout[k * 3 + 1] = v1;
  vout[k * 3 + 2] = v2;
}

// ---------------------------------------------------------------------------
// Kernel 2: boundary scan. 1024 threads; each folds 16 chunks with Q=M^128,
// then 10-step Kogge-Stone where the combine  w_t += M^(2^(11+s)) w_{t-2^s}
// is done 16 state-vectors at a time with v_wmma_f32_16x16x4_f32
// (A = power matrix, B = shifted vectors, C = current vectors).
// Emits all 16384 chunk-boundary states.
// ---------------------------------------------------------------------------
__global__ void __launch_bounds__(1024) k_scan(const float* __restrict__ vin,
                                               const float* __restrict__ pw,
                                               const float* __restrict__ x0,
                                               float* __restrict__ Sout) {
  __shared__ float W[2][1024][4];
  const int t = threadIdx.x;
  const int lane = t & 31;
  const int half = lane >> 4;
  const int l = lane & 15;
  const int waveBase = t & ~31;

  float Q[9];
#pragma unroll
  for (int i = 0; i < 9; ++i) Q[i] = pw[7 * 9 + i];  // Q = M^128

  float v[16][3];
#pragma unroll
  for (int i = 0; i < 16; ++i)
#pragma unroll
    for (int r = 0; r < 3; ++r) v[i][r] = vin[(16 * t + i) * 3 + r];

  // fold 16 chunk aggregates; thread 0 folds x0 in (w_init = x0)
  float w0 = 0.f, w1 = 0.f, w2 = 0.f;
  if (t == 0) { w0 = x0[0]; w1 = x0[1]; w2 = x0[2]; }
#pragma unroll
  for (int i = 0; i < 16; ++i) {
    const float n0 = fmaf(Q[0], w0, fmaf(Q[1], w1, fmaf(Q[2], w2, v[i][0])));
    const float n1 = fmaf(Q[3], w0, fmaf(Q[4], w1, fmaf(Q[5], w2, v[i][1])));
    const float n2 = fmaf(Q[6], w0, fmaf(Q[7], w1, fmaf(Q[8], w2, v[i][2])));
    w0 = n0; w1 = n1; w2 = n2;
  }
  W[0][t][0] = w0; W[0][t][1] = w1; W[0][t][2] = w2;
  __syncthreads();

  int src = 0;
  for (int s = 0; s < 10; ++s) {
    const float* R = pw + (11 + s) * 9;  // (Q^16)^(2^s) = M^(2^(11+s))
    v2f a; a[0] = 0.f; a[1] = 0.f;
    if (l < 3) {
      if (half == 0) { a[0] = R[l * 3 + 0]; a[1] = R[l * 3 + 1]; }
      else           { a[0] = R[l * 3 + 2]; }
    }
    const int dstep = 1 << s;
    float res[2][3];
#pragma unroll
    for (int g = 0; g < 2; ++g) {
      const int bcol = waveBase + g * 16 + l;
      const int prev = bcol - dstep;
      v2f b; b[0] = 0.f; b[1] = 0.f;
      if (prev >= 0) {
        if (half == 0) { b[0] = W[src][prev][0]; b[1] = W[src][prev][1]; }
        else           { b[0] = W[src][prev][2]; }
      }
      v8f c = {0.f, 0.f, 0.f, 0.f, 0.f, 0.f, 0.f, 0.f};
      if (half == 0) {
        c[0] = W[src][bcol][0];
        c[1] = W[src][bcol][1];
        c[2] = W[src][bcol][2];
      }
      v8f dmat = __builtin_amdgcn_wmma_f32_16x16x4_f32(false, a, false, b,
                                                       (short)0, c, false, false);
      res[g][0] = dmat[0]; res[g][1] = dmat[1]; res[g][2] = dmat[2];
    }
    __syncthreads();
    const int dst = src ^ 1;
    if (half == 0) {
#pragma unroll
      for (int g = 0; g < 2; ++g) {
        const int col = waveBase + g * 16 + l;
        W[dst][col][0] = res[g][0];
        W[dst][col][1] = res[g][1];
        W[dst][col][2] = res[g][2];
      }
    }
    __syncthreads();
    src ^= 1;
  }

  // expand to per-chunk boundary states
  float s0, s1, s2;
  if (t == 0) { s0 = x0[0]; s1 = x0[1]; s2 = x0[2]; }
  else { s0 = W[src][t - 1][0]; s1 = W[src][t - 1][1]; s2 = W[src][t - 1][2]; }
#pragma unroll
  for (int i = 0; i < 16; ++i) {
    Sout[(16 * t + i) * 3 + 0] = s0;
    Sout[(16 * t + i) * 3 + 1] = s1;
    Sout[(16 * t + i) * 3 + 2] = s2;
    const float n0 = fmaf(Q[0], s0, fmaf(Q[1], s1, fmaf(Q[2], s2, v[i][0])));
    const float n1 = fmaf(Q[3], s0, fmaf(Q[4], s1, fmaf(Q[5], s2, v[i][1])));
    const float n2 = fmaf(Q[6], s0, fmaf(Q[7], s1, fmaf(Q[8], s2, v[i][2])));
    s0 = n0; s1 = n1; s2 = n2;
  }
}

// ---------------------------------------------------------------------------
// Kernel 3: replay each chunk from its boundary state and write all outputs.
// d re-read hits L2 (192 MB >> 40 MB); writes the 24 MB result.
// ---------------------------------------------------------------------------
__global__ void __launch_bounds__(256) k_expand(const float* __restrict__ d,
                                                const float* __restrict__ pw,
                                                const float* __restrict__ params,
                                                const float* __restrict__ S,
                                                float* __restrict__ out,
                                                int nsteps) {
  const int k = blockIdx.x * blockDim.x + threadIdx.x;
  if (k >= NCHUNKS) return;
  float M[9];
#pragma unroll
  for (int i = 0; i < 9; ++i) M[i] = pw[i];
  const float Cai = params[0], Cwe = params[1], Cwi = params[2];
  const float Re = params[3], Rw = params[5], Rg = params[6];
  const float b00 = DT_F / (Cai * Rg), b01 = DT_F / Cai;
  const float b10 = DT_F / (Cwe * Re), b13 = DT_F / Cwe;
  const float b24 = DT_F / Cwi;

  float x0v = S[k * 3 + 0], x1v = S[k * 3 + 1], x2v = S[k * 3 + 2];
  if (k == 0) { out[0] = x0v; out[1] = x1v; out[2] = x2v; }  // row 0 = x0
  const long long base = (long long)k * CHUNK_LEN;
  for (int s = 0; s < CHUNK_LEN; ++s) {
    const long long j = base + s;
    if (j >= nsteps) break;
    const float* dr = d + (size_t)j * 5;
    __builtin_prefetch(dr + 40, 0, 0);
    const float d0 = dr[0], d1 = dr[1], d2 = dr[2], d3 = dr[3], d4 = dr[4];
    const float c0 = fmaf(b00, d0, b01 * (d1 + d2));
    const float c1 = fmaf(b10, d0, b13 * d3);
    const float c2 = b24 * d4;
    const float n0 = fmaf(M[0], x0v, fmaf(M[1], x1v, fmaf(M[2], x2v, c0)));
    const float n1 = fmaf(M[3], x0v, fmaf(M[4], x1v, fmaf(M[5], x2v, c1)));
    const float n2 = fmaf(M[6], x0v, fmaf(M[7], x1v, fmaf(M[8], x2v, c2)));
    x0v = n0; x1v = n1; x2v = n2;
    float* o = out + (size_t)(j + 1) * 3;
    o[0] = x0v; o[1] = x1v; o[2] = x2v;
  }
}

extern "C" void kernel_launch(void* const* d_in, const int* in_sizes, int n_in,
                              void* d_out, int out_size, void* d_ws, size_t ws_size,
                              hipStream_t stream) {
  (void)n_in; (void)out_size; (void)ws_size;
  const float* params = (const float*)d_in[0];
  const float* x0 = (const float*)d_in[1];
  const float* d = (const float*)d_in[2];
  float* out = (float*)d_out;
  const long long N = (long long)in_sizes[2] / 5;
  const int nsteps = (int)(N - 1);

  float* pw = (float*)d_ws;            // [21][9] matrix powers (pad to 256)
  float* v = pw + 256;                 // [NCHUNKS][3] chunk aggregates
  float* S = v + (size_t)NCHUNKS * 3;  // [NCHUNKS][3] boundary states

  k_init<<<1, 32, 0, stream>>>(params, pw);
  k_reduce<<<NCHUNKS / 256, 256, 0, stream>>>(d, pw, params, v, nsteps);
  k_scan<<<1, 1024, 0, stream>>>(v, pw, x0, S);
  k_expand<<<NCHUNKS / 256, 256, 0, stream>>>(d, pw, params, S, out, nsteps);
}